// FeatureOptimalLoss_1065151889711
// MI455X (gfx1250) — compile-verified
//
#include <hip/hip_runtime.h>
#include <hip/hip_bf16.h>

typedef __attribute__((ext_vector_type(2))) float v2f;
typedef __attribute__((ext_vector_type(8))) float v8f;

#define NN 2048
#define DIM 64
#define EPS 100.0f
#define INV_EPS 0.01f
#define ERR_TOL 0.1f
#define LOG_MU (-7.6246190f) /* log(1/2048) */

// ---------------------------------------------------------------------------
// init: zero u, v, u_new, v_new, done flag, and the scalar output
// ---------------------------------------------------------------------------
__global__ void k_init(float* u, float* v, float* un, float* vn, int* done, float* out) {
    int i = blockIdx.x * blockDim.x + threadIdx.x;
    if (i < NN) { u[i] = 0.0f; v[i] = 0.0f; un[i] = 0.0f; vn[i] = 0.0f; }
    if (i == 0) { *done = 0; out[0] = 0.0f; }
}

// ---------------------------------------------------------------------------
// pairwise L1: out[x][y] = sum_d |A[x][d] - B[y][d]|   (out is NN x NN)
// 32x32 tile per 256-thread block, operands staged in LDS via float4 loads.
// Called twice with swapped operands to produce both C and C^T coalesced.
// ---------------------------------------------------------------------------
__global__ void __launch_bounds__(256) k_pairwise_l1(const float* __restrict__ A,
                                                     const float* __restrict__ B,
                                                     float* __restrict__ out) {
    __shared__ __align__(16) float As[32 * DIM];
    __shared__ __align__(16) float Bs[32 * DIM];
    const int tid = threadIdx.x;
    const int xb = blockIdx.x * 32;
    const int yb = blockIdx.y * 32;

    #pragma unroll
    for (int rep = 0; rep < 2; ++rep) {
        int idx = tid + rep * 256;            // float4 index in [0,512)
        int row = idx >> 4;                   // [0,32)
        int c4  = (idx & 15) << 2;            // [0,64) step 4
        *reinterpret_cast<float4*>(&As[row * DIM + c4]) =
            *reinterpret_cast<const float4*>(A + (size_t)(xb + row) * DIM + c4);
        *reinterpret_cast<float4*>(&Bs[row * DIM + c4]) =
            *reinterpret_cast<const float4*>(B + (size_t)(yb + row) * DIM + c4);
    }
    __syncthreads();

    const int r  = tid >> 3;        // x-local row  [0,32)
    const int cb = (tid & 7) << 2;  // y-local base {0,4,...,28}
    float s0 = 0.f, s1 = 0.f, s2 = 0.f, s3 = 0.f;
    #pragma unroll 8
    for (int d = 0; d < DIM; ++d) {
        float av = As[r * DIM + d];
        s0 += fabsf(av - Bs[(cb + 0) * DIM + d]);
        s1 += fabsf(av - Bs[(cb + 1) * DIM + d]);
        s2 += fabsf(av - Bs[(cb + 2) * DIM + d]);
        s3 += fabsf(av - Bs[(cb + 3) * DIM + d]);
    }
    float4 o = make_float4(s0, s1, s2, s3);
    *reinterpret_cast<float4*>(out + (size_t)(xb + r) * NN + yb + cb) = o;
}

// ---------------------------------------------------------------------------
// LSE reduction via WMMA: for each n, s_n = sum_k exp((addn[n]+addk[k]-Mtx[k][n])/eps)
// outn[n] = eps*(log_mu - log(s_n + 1e-6)) + basen[n]
// One wave owns 16 columns n; K runs over all 2048 rows in chunks of 4 fed to
// V_WMMA_F32_16X16X4_F32 with A = ones (sum over K is K-layout invariant).
// Mtx is read K-major: Mtx[k*NN + n] -> coalesced across lanes.
// ---------------------------------------------------------------------------
__global__ void __launch_bounds__(256) k_lse(const float* __restrict__ Mtx,
                                             const float* __restrict__ addk,
                                             const float* __restrict__ addn,
                                             const float* __restrict__ basen,
                                             float* __restrict__ outn) {
    const int lane  = threadIdx.x & 31;
    const int wave  = threadIdx.x >> 5;
    const int n     = (blockIdx.x * 8 + wave) * 16 + (lane & 15);
    const int half2 = (lane >> 4) << 1;   // 0 or 2
    const float an = addn[n];

    v8f acc = {0.f, 0.f, 0.f, 0.f, 0.f, 0.f, 0.f, 0.f};
    v2f a;  a.x = 1.0f; a.y = 1.0f;       // ones matrix (A, 16x4)

    for (int k = 0; k < NN; k += 4) {
        int k0 = k + half2;
        float m0 = Mtx[(size_t)k0 * NN + n];
        float m1 = Mtx[(size_t)(k0 + 1) * NN + n];
        v2f b;
        b.x = __expf((an + addk[k0]     - m0) * INV_EPS);
        b.y = __expf((an + addk[k0 + 1] - m1) * INV_EPS);
        acc = __builtin_amdgcn_wmma_f32_16x16x4_f32(false, a, false, b,
                                                    (short)0, acc, false, false);
    }
    if (lane < 16) {
        float s = acc[0];   // D[0][n] = column sum for n
        outn[n] = EPS * (LOG_MU - __logf(s + 1e-6f)) + basen[n];
    }
}

// Fallback (no C^T available): reads Mtx[n*NN + k] (transposed access).
__global__ void __launch_bounds__(256) k_lse_t(const float* __restrict__ Mtx,
                                               const float* __restrict__ addk,
                                               const float* __restrict__ addn,
                                               const float* __restrict__ basen,
                                               float* __restrict__ outn) {
    const int lane  = threadIdx.x & 31;
    const int wave  = threadIdx.x >> 5;
    const int n     = (blockIdx.x * 8 + wave) * 16 + (lane & 15);
    const int half2 = (lane >> 4) << 1;
    const float an = addn[n];

    v8f acc = {0.f, 0.f, 0.f, 0.f, 0.f, 0.f, 0.f, 0.f};
    v2f a;  a.x = 1.0f; a.y = 1.0f;

    for (int k = 0; k < NN; k += 4) {
        int k0 = k + half2;
        float m0 = Mtx[(size_t)n * NN + k0];
        float m1 = Mtx[(size_t)n * NN + k0 + 1];
        v2f b;
        b.x = __expf((an + addk[k0]     - m0) * INV_EPS);
        b.y = __expf((an + addk[k0 + 1] - m1) * INV_EPS);
        acc = __builtin_amdgcn_wmma_f32_16x16x4_f32(false, a, false, b,
                                                    (short)0, acc, false, false);
    }
    if (lane < 16) {
        float s = acc[0];
        outn[n] = EPS * (LOG_MU - __logf(s + 1e-6f)) + basen[n];
    }
}

// ---------------------------------------------------------------------------
// commit: err = sum|u_new - u|; gated by previous done flag (reference semantics)
// ---------------------------------------------------------------------------
__global__ void __launch_bounds__(1024) k_commit(float* __restrict__ u,
                                                 float* __restrict__ v,
                                                 const float* __restrict__ un,
                                                 const float* __restrict__ vn,
                                                 int* __restrict__ done) {
    __shared__ float red[1024];
    const int t = threadIdx.x;
    const int dold = *done;

    float u0 = u[t], u1 = u[t + 1024];
    float n0 = un[t], n1 = un[t + 1024];
    red[t] = fabsf(n0 - u0) + fabsf(n1 - u1);
    __syncthreads();
    #pragma unroll
    for (int s = 512; s > 0; s >>= 1) {
        if (t < s) red[t] += red[t + s];
        __syncthreads();
    }
    if (!dold) {
        u[t] = n0; u[t + 1024] = n1;
        v[t] = vn[t]; v[t + 1024] = vn[t + 1024];
    }
    if (t == 0) *done = dold | (red[0] < ERR_TOL ? 1 : 0);
}

// ---------------------------------------------------------------------------
// final: sum_ij exp((-C[i][j]+u[i]+v[j])/eps) * C[i][j]
// Same WMMA reduction trick (k=i over rows, n=j columns), then wave reduce.
// ---------------------------------------------------------------------------
__global__ void __launch_bounds__(256) k_final(const float* __restrict__ C,
                                               const float* __restrict__ u,
                                               const float* __restrict__ v,
                                               float* __restrict__ out) {
    const int lane  = threadIdx.x & 31;
    const int wave  = threadIdx.x >> 5;
    const int n     = (blockIdx.x * 8 + wave) * 16 + (lane & 15);
    const int half2 = (lane >> 4) << 1;
    const float vn_ = v[n];

    v8f acc = {0.f, 0.f, 0.f, 0.f, 0.f, 0.f, 0.f, 0.f};
    v2f a;  a.x = 1.0f; a.y = 1.0f;

    for (int k = 0; k < NN; k += 4) {
        int k0 = k + half2;
        float c0 = C[(size_t)k0 * NN + n];
        float c1 = C[(size_t)(k0 + 1) * NN + n];
        v2f b;
        b.x = __expf((vn_ + u[k0]     - c0) * INV_EPS) * c0;
        b.y = __expf((vn_ + u[k0 + 1] - c1) * INV_EPS) * c1;
        acc = __builtin_amdgcn_wmma_f32_16x16x4_f32(false, a, false, b,
                                                    (short)0, acc, false, false);
    }
    float s = (lane < 16) ? acc[0] : 0.0f;   // lanes 16-31 duplicate M=8 row
    #pragma unroll
    for (int off = 16; off > 0; off >>= 1) s += __shfl_down(s, off, 32);
    if (lane == 0) atomicAdd(out, s);
}

// ---------------------------------------------------------------------------
extern "C" void kernel_launch(void* const* d_in, const int* in_sizes, int n_in,
                              void* d_out, int out_size, void* d_ws, size_t ws_size,
                              hipStream_t stream) {
    (void)in_sizes; (void)n_in; (void)out_size;
    const float* outp = (const float*)d_in[0];   // "output" [2048,64]
    const float* targ = (const float*)d_in[1];   // "target" [2048,64]
    float* out = (float*)d_out;
    float* ws  = (float*)d_ws;

    const size_t plane = (size_t)NN * NN;
    const size_t vecBytes = 4u * NN * sizeof(float) + sizeof(int) + 64;
    const bool haveCT = ws_size >= 2 * plane * sizeof(float) + vecBytes;

    float* C  = ws;
    float* CT = haveCT ? (ws + plane) : nullptr;
    float* u  = ws + (haveCT ? 2 : 1) * plane;
    float* v  = u + NN;
    float* un = v + NN;
    float* vn = un + NN;
    int*  done = (int*)(vn + NN);

    k_init<<<8, 256, 0, stream>>>(u, v, un, vn, done, out);

    dim3 g(NN / 32, NN / 32);
    k_pairwise_l1<<<g, 256, 0, stream>>>(outp, targ, C);       // C[i][j]
    if (haveCT)
        k_pairwise_l1<<<g, 256, 0, stream>>>(targ, outp, CT);  // CT[j][i] = C[i][j]

    const int lseBlocks = NN / (16 * 8);  // 16 blocks of 8 waves x 16 cols
    for (int it = 0; it < 10; ++it) {
        // u_new: row-LSE (sum over j) -> read CT coalesced (or C transposed)
        if (haveCT)
            k_lse  <<<lseBlocks, 256, 0, stream>>>(CT, v, u, u, un);
        else
            k_lse_t<<<lseBlocks, 256, 0, stream>>>(C,  v, u, u, un);
        // v_new: col-LSE (sum over i) using u_new -> read C coalesced
        k_lse<<<lseBlocks, 256, 0, stream>>>(C, un, v, v, vn);
        k_commit<<<1, 1024, 0, stream>>>(u, v, un, vn, done);
    }

    k_final<<<lseBlocks, 256, 0, stream>>>(C, u, v, out);
}